// GNNEncoder_3092376453139
// MI455X (gfx1250) — compile-verified
//
#include <hip/hip_runtime.h>

typedef __attribute__((ext_vector_type(2))) float v2f;
typedef __attribute__((ext_vector_type(8))) float v8f;

#define N_NODES 100000
#define N_EDGES 1600000
#define IN_DIM  128
#define HID_DIM 64
#define OUT_DIM 16
#define BN_EPS  1e-5f

// ---------------------------------------------------------------- zero
__global__ void zero_kernel(float* __restrict__ p, long n) {
    long i = (long)blockIdx.x * blockDim.x + threadIdx.x;
    long stride = (long)gridDim.x * blockDim.x;
    for (; i < n; i += stride) p[i] = 0.0f;
}

// ---------------------------------------------------------------- degree (cnt[dst] += 1)
__global__ void degree_kernel(const int* __restrict__ ei, float* __restrict__ cnt) {
    int e = blockIdx.x * blockDim.x + threadIdx.x;
    if (e < N_EDGES) atomicAdd(&cnt[ei[N_EDGES + e]], 1.0f);
}

// ---------------------------------------------------------------- edge scatter: sums[dst] += feat[src]
// One edge per 32-lane group; each lane handles PER_LANE consecutive floats.
template <int F, int PER_LANE>
__global__ void scatter_kernel(const float* __restrict__ feat,
                               const int* __restrict__ ei,
                               float* __restrict__ sums) {
    long t = (long)blockIdx.x * blockDim.x + threadIdx.x;
    long e = t >> 5;
    int lane = (int)(t & 31);
    if (e >= N_EDGES) return;
    int src = ei[e];
    int dst = ei[N_EDGES + e];
    const float* xs = feat + (long)src * F + lane * PER_LANE;
    float* so       = sums + (long)dst * F + lane * PER_LANE;
    float v[PER_LANE];
#pragma unroll
    for (int j = 0; j < PER_LANE; ++j) v[j] = xs[j];
#pragma unroll
    for (int j = 0; j < PER_LANE; ++j) atomicAdd(&so[j], v[j]);
}

// ---------------------------------------------------------------- layer-1 GEMM via WMMA
// h[16x64 tile] = (sums/cnt) @ W1l^T + b1 + x @ W1r^T ; also accumulate BN sum/sumsq.
__global__ __launch_bounds__(256)
void gemm1_kernel(const float* __restrict__ x,
                  const float* __restrict__ sums,
                  const float* __restrict__ cnt,
                  const float* __restrict__ W1l,
                  const float* __restrict__ b1,
                  const float* __restrict__ W1r,
                  float* __restrict__ h,
                  float* __restrict__ bnstats) {
    __shared__ float sW[2 * HID_DIM * IN_DIM]; // 64 KB: W1l then W1r, row-major [64][128]
    for (int i = threadIdx.x; i < HID_DIM * IN_DIM; i += blockDim.x) {
        sW[i]                    = W1l[i];
        sW[HID_DIM * IN_DIM + i] = W1r[i];
    }
    __syncthreads();

    int wave = threadIdx.x >> 5;
    int lane = threadIdx.x & 31;
    int tile = blockIdx.x * 8 + wave;          // wave-uniform
    if (tile >= N_NODES / 16) return;

    int m0   = tile * 16;
    int half = lane >> 4;                      // 0: K+{0,1}, 1: K+{2,3}
    int lr   = lane & 15;                      // A row-in-tile / B column
    int row  = m0 + lr;

    float inv = 1.0f / fmaxf(cnt[row], 1.0f);
    const float* aggRow = sums + (long)row * IN_DIM;
    const float* xRow   = x    + (long)row * IN_DIM;

    v8f acc[4] = {};                           // 4 N-tiles of 16 cols each

    for (int k0 = 0; k0 < IN_DIM; k0 += 4) {
        int ka = k0 + half * 2;
        v2f aAgg = { aggRow[ka] * inv, aggRow[ka + 1] * inv };
        v2f aX   = { xRow[ka],         xRow[ka + 1] };
#pragma unroll
        for (int t = 0; t < 4; ++t) {
            int n = t * 16 + lr;
            const float* wl = sW + n * IN_DIM + ka;
            const float* wr = sW + HID_DIM * IN_DIM + n * IN_DIM + ka;
            v2f bL = { wl[0], wl[1] };
            v2f bR = { wr[0], wr[1] };
            acc[t] = __builtin_amdgcn_wmma_f32_16x16x4_f32(
                false, aAgg, false, bL, (short)0, acc[t], false, false);
            acc[t] = __builtin_amdgcn_wmma_f32_16x16x4_f32(
                false, aX,   false, bR, (short)0, acc[t], false, false);
        }
    }

    // epilogue: bias, store pre-BN h, accumulate BN statistics
#pragma unroll
    for (int t = 0; t < 4; ++t) {
        int n = t * 16 + lr;
        float bias = b1[n];
        float s = 0.0f, sq = 0.0f;
#pragma unroll
        for (int v = 0; v < 8; ++v) {
            float val = acc[t][v] + bias;
            int m = m0 + half * 8 + v;
            h[(long)m * HID_DIM + n] = val;
            s  += val;
            sq += val * val;
        }
        atomicAdd(&bnstats[n], s);
        atomicAdd(&bnstats[HID_DIM + n], sq);
    }
}

// ---------------------------------------------------------------- BN (training stats) + ReLU, in place
__global__ void bn_relu_kernel(float* __restrict__ h,
                               const float* __restrict__ bnstats,
                               const float* __restrict__ gamma,
                               const float* __restrict__ beta) {
    long i = (long)blockIdx.x * blockDim.x + threadIdx.x;
    if (i >= (long)N_NODES * HID_DIM) return;
    int c = (int)(i & (HID_DIM - 1));
    const float invN = 1.0f / (float)N_NODES;
    float mean = bnstats[c] * invN;
    float var  = bnstats[HID_DIM + c] * invN - mean * mean;
    float val  = (h[i] - mean) * rsqrtf(var + BN_EPS) * gamma[c] + beta[c];
    h[i] = fmaxf(val, 0.0f);
}

// ---------------------------------------------------------------- layer-2 GEMM via WMMA
// out[16x16 tile] = (sums/cnt) @ W2l^T + b2 + h @ W2r^T
__global__ __launch_bounds__(256)
void gemm2_kernel(const float* __restrict__ h,
                  const float* __restrict__ sums,
                  const float* __restrict__ cnt,
                  const float* __restrict__ W2l,
                  const float* __restrict__ b2,
                  const float* __restrict__ W2r,
                  float* __restrict__ out) {
    __shared__ float sW[2 * OUT_DIM * HID_DIM]; // 8 KB
    for (int i = threadIdx.x; i < OUT_DIM * HID_DIM; i += blockDim.x) {
        sW[i]                     = W2l[i];
        sW[OUT_DIM * HID_DIM + i] = W2r[i];
    }
    __syncthreads();

    int wave = threadIdx.x >> 5;
    int lane = threadIdx.x & 31;
    int tile = blockIdx.x * 8 + wave;
    if (tile >= N_NODES / 16) return;

    int m0   = tile * 16;
    int half = lane >> 4;
    int lr   = lane & 15;
    int row  = m0 + lr;

    float inv = 1.0f / fmaxf(cnt[row], 1.0f);
    const float* aggRow = sums + (long)row * HID_DIM;
    const float* hRow   = h    + (long)row * HID_DIM;

    v8f acc = {};
    for (int k0 = 0; k0 < HID_DIM; k0 += 4) {
        int ka = k0 + half * 2;
        v2f aAgg = { aggRow[ka] * inv, aggRow[ka + 1] * inv };
        v2f aH   = { hRow[ka],         hRow[ka + 1] };
        const float* wl = sW + lr * HID_DIM + ka;
        const float* wr = sW + OUT_DIM * HID_DIM + lr * HID_DIM + ka;
        v2f bL = { wl[0], wl[1] };
        v2f bR = { wr[0], wr[1] };
        acc = __builtin_amdgcn_wmma_f32_16x16x4_f32(
            false, aAgg, false, bL, (short)0, acc, false, false);
        acc = __builtin_amdgcn_wmma_f32_16x16x4_f32(
            false, aH,   false, bR, (short)0, acc, false, false);
    }

    float bias = b2[lr];
#pragma unroll
    for (int v = 0; v < 8; ++v) {
        int m = m0 + half * 8 + v;
        out[(long)m * OUT_DIM + lr] = acc[v] + bias;
    }
}

// ---------------------------------------------------------------- launch
extern "C" void kernel_launch(void* const* d_in, const int* in_sizes, int n_in,
                              void* d_out, int out_size, void* d_ws, size_t ws_size,
                              hipStream_t stream) {
    const float* x     = (const float*)d_in[0];
    const int*   ei    = (const int*)  d_in[1];
    const float* W1l   = (const float*)d_in[2];
    const float* b1    = (const float*)d_in[3];
    const float* W1r   = (const float*)d_in[4];
    const float* gamma = (const float*)d_in[5];
    const float* beta  = (const float*)d_in[6];
    const float* W2l   = (const float*)d_in[7];
    const float* b2    = (const float*)d_in[8];
    const float* W2r   = (const float*)d_in[9];
    float* out = (float*)d_out;

    // workspace layout (floats): sums[N*128] | cnt[N] | bnstats[128] | h[N*64]
    float* ws      = (float*)d_ws;
    float* sums    = ws;
    float* cnt     = sums + (size_t)N_NODES * IN_DIM;
    float* bnstats = cnt + N_NODES;
    float* h       = bnstats + 2 * HID_DIM;

    const int ntiles = N_NODES / 16;                 // 6250 (exact)
    const long scatterThreads = (long)N_EDGES * 32;  // one 32-lane group per edge

    // zero sums1 + cnt + bnstats (contiguous)
    zero_kernel<<<2048, 256, 0, stream>>>(sums, (long)N_NODES * IN_DIM + N_NODES + 2 * HID_DIM);
    // degrees
    degree_kernel<<<(N_EDGES + 255) / 256, 256, 0, stream>>>(ei, cnt);
    // layer-1 segment sum: sums[dst][0:128] += x[src]
    scatter_kernel<IN_DIM, 4><<<(int)((scatterThreads + 255) / 256), 256, 0, stream>>>(x, ei, sums);
    // layer-1 dense part + BN stats
    gemm1_kernel<<<(ntiles + 7) / 8, 256, 0, stream>>>(x, sums, cnt, W1l, b1, W1r, h, bnstats);
    // BN + ReLU
    bn_relu_kernel<<<(int)(((long)N_NODES * HID_DIM + 255) / 256), 256, 0, stream>>>(h, bnstats, gamma, beta);
    // reuse sums buffer for layer-2 segment sums (first N*64 floats)
    zero_kernel<<<2048, 256, 0, stream>>>(sums, (long)N_NODES * HID_DIM);
    scatter_kernel<HID_DIM, 2><<<(int)((scatterThreads + 255) / 256), 256, 0, stream>>>(h, ei, sums);
    // layer-2 dense part -> output
    gemm2_kernel<<<(ntiles + 7) / 8, 256, 0, stream>>>(h, sums, cnt, W2l, b2, W2r, out);
}